// RC_30966714204178
// MI455X (gfx1250) — compile-verified
//
#include <hip/hip_runtime.h>
#include <hip/hip_bf16.h>
#include <stdint.h>

// Problem constants (from reference setup_inputs)
#define BB  256
#define TT  250
#define IN  700
#define HH  128
#define OUTN 20
#define BT  (BB * TT)          // 64000

typedef __attribute__((ext_vector_type(2))) float v2f;
typedef __attribute__((ext_vector_type(8))) float v8f;

// Async memory->LDS copy (CDNA5 GLOBAL_LOAD_ASYNC_TO_LDS_B128, ASYNCcnt).
// lds_off is the wave-relative LDS byte offset (low 32 bits of the flat addr).
__device__ __forceinline__ void async_load_b128(uint32_t lds_off, const void* gaddr) {
    asm volatile("global_load_async_to_lds_b128 %0, %1, off"
                 :: "v"(lds_off), "v"(gaddr) : "memory");
}
__device__ __forceinline__ void wait_async0() {
    asm volatile("s_wait_asynccnt 0x0" ::: "memory");
}

// ---------------------------------------------------------------------------
// Kernel 1: fused  d0 = x @ W_in^T + b_in  followed by 4 chained
//           d_{l+1} = d_l @ A_w[l]^T + A_b[l], storing each layer to drives.
// One block = one 16-row M-tile of [B*T]; 8 waves, one per 16-wide N-tile of
// H=128; fp32 WMMA 16x16x4 with software-pipelined fragment loads.
// ---------------------------------------------------------------------------
__global__ __launch_bounds__(256) void snn_drives_kernel(
    const float* __restrict__ x, const float* __restrict__ W_in,
    const float* __restrict__ b_in, const float* __restrict__ A_w,
    const float* __restrict__ A_b, float* __restrict__ drives)
{
    __shared__ __align__(16) float lds_x[16 * IN];   // 44800 B: 16 rows of x
    __shared__ float dbuf0[16 * HH];                 // 8192 B
    __shared__ float dbuf1[16 * HH];                 // 8192 B

    const int tid  = threadIdx.x;
    const int wave = tid >> 5;
    const int lane = tid & 31;
    const int m0   = blockIdx.x * 16;                // BT = 4000 * 16

    // Stage the 16 x 700 x-tile straight into LDS via async DMA loads
    // (2800 aligned b128 transfers; row stride 2800 B is 16B-aligned).
    {
        const uint32_t lds_base = (uint32_t)(uintptr_t)lds_x;
        for (int i = tid; i < (16 * IN) / 4; i += 256) {
            const int r  = i / (IN / 4);
            const int c4 = i - r * (IN / 4);
            const float* g = x + (size_t)(m0 + r) * IN + c4 * 4;
            async_load_b128(lds_base + (uint32_t)(r * IN + c4 * 4) * 4u, g);
        }
        wait_async0();      // own ASYNCcnt drained before signaling barrier
    }
    __syncthreads();

    const int lm   = lane & 15;            // A sub-row / B,C,D sub-col
    const int kh   = (lane >> 4) << 1;     // K phase: 0 (lanes 0-15) or 2
    const int mrow = (lane >> 4) << 3;     // C/D row base for half-wave
    const int n0   = wave * 16;

    // ---- GEMM 1: K = 700 (175 k-steps of 4), pipelined ---------------------
    {
        const float* wrow = W_in + (size_t)(n0 + lm) * IN;
        const float* arow = lds_x + lm * IN;

        // Pull this wave's B row toward the caches up front; no branches in
        // the hot loop (each lane owns a distinct W_in row).
        #pragma unroll
        for (int p = 0; p < 7; ++p)
            __builtin_prefetch(wrow + p * 100, 0, 1);   // global_prefetch_b8

        v8f c = {};
        v2f av, bv;
        av.x = arow[kh];     av.y = arow[kh + 1];
        bv.x = wrow[kh];     bv.y = wrow[kh + 1];
        #pragma unroll 8
        for (int k = 0; k < IN - 4; k += 4) {
            v2f an, bn;                                  // prefetch next frags
            an.x = arow[k + 4 + kh]; an.y = arow[k + 4 + kh + 1];
            bn.x = wrow[k + 4 + kh]; bn.y = wrow[k + 4 + kh + 1];
            c = __builtin_amdgcn_wmma_f32_16x16x4_f32(
                    false, av, false, bv, (short)0, c, false, false);
            av = an; bv = bn;
        }
        c = __builtin_amdgcn_wmma_f32_16x16x4_f32(
                false, av, false, bv, (short)0, c, false, false);

        const float bias = b_in[n0 + lm];
        #pragma unroll
        for (int v = 0; v < 8; ++v)
            dbuf0[(mrow + v) * HH + n0 + lm] = c[v] + bias;
    }
    __syncthreads();

    // ---- Chained layers: K = 128 (32 k-steps of 4), pipelined --------------
    float* src = dbuf0;
    float* dst = dbuf1;
    for (int l = 0; l < 4; ++l) {
        const float* wrow = A_w + ((size_t)l * HH + (n0 + lm)) * HH;
        const float* arow = src + lm * HH;

        v8f c = {};
        v2f av, bv;
        av.x = arow[kh];     av.y = arow[kh + 1];
        bv.x = wrow[kh];     bv.y = wrow[kh + 1];
        #pragma unroll 8
        for (int k = 0; k < HH - 4; k += 4) {
            v2f an, bn;
            an.x = arow[k + 4 + kh]; an.y = arow[k + 4 + kh + 1];
            bn.x = wrow[k + 4 + kh]; bn.y = wrow[k + 4 + kh + 1];
            c = __builtin_amdgcn_wmma_f32_16x16x4_f32(
                    false, av, false, bv, (short)0, c, false, false);
            av = an; bv = bn;
        }
        c = __builtin_amdgcn_wmma_f32_16x16x4_f32(
                false, av, false, bv, (short)0, c, false, false);

        const float bias = A_b[l * HH + n0 + lm];
        float* gout = drives + ((size_t)l * BT + m0) * HH;
        #pragma unroll
        for (int v = 0; v < 8; ++v) {
            const float val = c[v] + bias;
            dst[(mrow + v) * HH + n0 + lm] = val;
            gout[(size_t)(mrow + v) * HH + n0 + lm] = val;
        }
        __syncthreads();
        float* tmp = src; src = dst; dst = tmp;
    }
}

// ---------------------------------------------------------------------------
// Kernel 2: elementwise LIF scan. One thread per (l, b, h) sequence; lanes map
// to consecutive h so each per-t drives read is a contiguous 128B line/wave.
// drives is read-once and hid4 is write-once: use non-temporal (TH=NT).
// ---------------------------------------------------------------------------
__global__ __launch_bounds__(256) void snn_scan_kernel(
    const float* __restrict__ drives, const float* __restrict__ mem0,
    const float* __restrict__ thr, const float* __restrict__ decay,
    const float* __restrict__ rst, float* __restrict__ feat,
    float* __restrict__ hid_mem, float* __restrict__ hid_spk)
{
    const int idx = blockIdx.x * blockDim.x + threadIdx.x;  // 0 .. 4*B*H-1
    const int h = idx & (HH - 1);
    const int b = (idx >> 7) & (BB - 1);
    const int l = idx >> 15;           // B*H = 32768 = 2^15

    const float th = thr[h], dc = decay[h], rs = rst[h];
    const float m0v = mem0[(size_t)l * BB * HH + (size_t)b * HH + h];
    float mem = m0v, spk = 0.f, ssum = 0.f, msum = 0.f;

    const float* dptr = drives + ((size_t)l * BT + (size_t)b * TT) * HH + h;
    float* hm = hid_mem + (size_t)b * (TT + 1) * HH + h;
    float* hs = hid_spk + (size_t)b * (TT + 1) * HH + h;
    const bool rec = (l == 3);
    if (rec) {
        __builtin_nontemporal_store(m0v, hm);
        __builtin_nontemporal_store(0.f, hs);
    }

    #pragma unroll 4
    for (int t = 0; t < TT; ++t) {
        const float drive = __builtin_nontemporal_load(dptr + (size_t)t * HH);
        mem = rs * spk + mem * dc * (1.f - spk) + drive;
        spk = (mem - th > 0.f) ? 1.f : 0.f;
        ssum += spk;
        msum += mem;
        if (rec) {
            __builtin_nontemporal_store(mem, hm + (size_t)(t + 1) * HH);
            __builtin_nontemporal_store(spk, hs + (size_t)(t + 1) * HH);
        }
    }
    feat[(size_t)b * (8 * HH) + l * HH + h] = ssum * (1.f / (float)TT);
    feat[(size_t)b * (8 * HH) + 4 * HH + l * HH + h] =
        (m0v + msum) * (1.f / (float)(TT + 1));
}

// ---------------------------------------------------------------------------
// Kernel 3: head GEMM out = feat @ W_out^T + b_out   [256,1024] x [20,1024]
// ---------------------------------------------------------------------------
__global__ __launch_bounds__(256) void snn_head_kernel(
    const float* __restrict__ feat, const float* __restrict__ W_out,
    const float* __restrict__ b_out, float* __restrict__ out)
{
    const int idx = blockIdx.x * blockDim.x + threadIdx.x;
    if (idx >= BB * OUTN) return;
    const int b = idx / OUTN, o = idx - b * OUTN;
    const float* f = feat + (size_t)b * (8 * HH);
    const float* w = W_out + (size_t)o * (8 * HH);
    float s = b_out[o];
    for (int k = 0; k < 8 * HH; ++k) s = fmaf(f[k], w[k], s);
    out[idx] = s;
}

// ---------------------------------------------------------------------------
// Kernel 4: A_norm = sum |A_w| over 4*128*128 elements, single-block reduce.
// ---------------------------------------------------------------------------
__global__ __launch_bounds__(256) void snn_norm_kernel(
    const float* __restrict__ A_w, float* __restrict__ out)
{
    __shared__ float red[256];
    float s = 0.f;
    for (int i = threadIdx.x; i < 4 * HH * HH; i += 256) s += fabsf(A_w[i]);
    red[threadIdx.x] = s;
    __syncthreads();
    for (int off = 128; off > 0; off >>= 1) {
        if ((int)threadIdx.x < off) red[threadIdx.x] += red[threadIdx.x + off];
        __syncthreads();
    }
    if (threadIdx.x == 0) out[0] = red[0];
}

// ---------------------------------------------------------------------------
extern "C" void kernel_launch(void* const* d_in, const int* in_sizes, int n_in,
                              void* d_out, int out_size, void* d_ws, size_t ws_size,
                              hipStream_t stream) {
    (void)in_sizes; (void)n_in; (void)out_size; (void)ws_size;
    const float* x     = (const float*)d_in[0];
    const float* W_in  = (const float*)d_in[1];
    const float* b_in  = (const float*)d_in[2];
    const float* A_w   = (const float*)d_in[3];
    const float* A_b   = (const float*)d_in[4];
    const float* W_out = (const float*)d_in[5];
    const float* b_out = (const float*)d_in[6];
    const float* thr   = (const float*)d_in[7];
    const float* decay = (const float*)d_in[8];
    const float* rst   = (const float*)d_in[9];
    const float* mem0  = (const float*)d_in[10];

    // Output tuple layout: out[256*20] | hid4_mem[256*251*128] |
    //                      hid4_spk[256*251*128] | A_norm[1]
    float* out      = (float*)d_out;
    float* hid_mem  = out + BB * OUTN;
    float* hid_spk  = hid_mem + (size_t)BB * (TT + 1) * HH;
    float* a_norm   = hid_spk + (size_t)BB * (TT + 1) * HH;

    // Workspace: drives [4, BT, H] (131 MB) + feat [B, 8H] (1 MB)
    float* drives = (float*)d_ws;
    float* feat   = drives + (size_t)4 * BT * HH;

    snn_drives_kernel<<<BT / 16, 256, 0, stream>>>(x, W_in, b_in, A_w, A_b, drives);
    snn_scan_kernel<<<(4 * BB * HH) / 256, 256, 0, stream>>>(
        drives, mem0, thr, decay, rst, feat, hid_mem, hid_spk);
    snn_head_kernel<<<(BB * OUTN + 255) / 256, 256, 0, stream>>>(
        feat, W_out, b_out, out);
    snn_norm_kernel<<<1, 256, 0, stream>>>(A_w, a_norm);
}